// RNN_70635032150757
// MI455X (gfx1250) — compile-verified
//
#include <hip/hip_runtime.h>
#include <hip/hip_bf16.h>
#include <math.h>

typedef __attribute__((ext_vector_type(2))) float    v2f;
typedef __attribute__((ext_vector_type(8))) float    v8f;
typedef __attribute__((ext_vector_type(4))) int      v4i;
typedef __attribute__((ext_vector_type(4))) unsigned v4u;
typedef __attribute__((ext_vector_type(8))) int      v8i;

#define VOCAB 50000
#define EDIM  300
#define HDIM  512
#define ODIM  4
#define BDIM  128
#define TDIM  512

// LDS row stride for the staged W_hh slice: 516 % 64 banks == 4, so the 16
// fragment lanes (rows) hit distinct banks instead of stride-512's 16-way clash.
#define WPAD  516

#if __has_builtin(__builtin_amdgcn_global_load_async_to_lds_b128)
#define HAVE_ASYNC_LDS 1
#else
#define HAVE_ASYNC_LDS 0
#endif

#if __has_builtin(__builtin_amdgcn_tensor_load_to_lds)
#define HAVE_TDM 1
#else
#define HAVE_TDM 0
#endif

// ---------------------------------------------------------------------------
// init: zero h0 buffer and the grid-barrier counter (must happen every launch)
// ---------------------------------------------------------------------------
__global__ void rnn_init_kernel(float* __restrict__ h0, unsigned* __restrict__ counter, int n) {
  int i = blockIdx.x * blockDim.x + threadIdx.x;
  if (i < n) h0[i] = 0.0f;
  if (i == 0) *counter = 0u;
}

// ---------------------------------------------------------------------------
// Kernel 1: embedding gather + input projection, fp32 WMMA 16x16x4.
//   xp[t][b][h] = sum_e emb[x[b*T+t]][e] * W_ih[h][e] + b_ih[h] + b_hh[h]
// ---------------------------------------------------------------------------
__global__ __launch_bounds__(128)
void embed_proj_kernel(const int* __restrict__ x,
                       const float* __restrict__ emb,
                       const float* __restrict__ W_ih,
                       const float* __restrict__ b_ih,
                       const float* __restrict__ b_hh,
                       float* __restrict__ xp) {
  const int lane = threadIdx.x & 31;
  const int wave = threadIdx.x >> 5;
  const int l    = lane & 15;
  const int half = lane >> 4;           // 0: K=k0..k0+1 | 1: K=k0+2..k0+3

  const int m0 = blockIdx.x * 64 + wave * 16;   // row-tile base (flat b*T+t)
  const int n0 = blockIdx.y * 64;               // col-tile base in H

  const int tok = x[m0 + l];
  const float* arow = emb + (size_t)tok * EDIM + 2 * half;

  v8f acc0 = {}, acc1 = {}, acc2 = {}, acc3 = {};

  const float* b0p = W_ih + (size_t)(n0 +  0 + l) * EDIM + 2 * half;
  const float* b1p = W_ih + (size_t)(n0 + 16 + l) * EDIM + 2 * half;
  const float* b2p = W_ih + (size_t)(n0 + 32 + l) * EDIM + 2 * half;
  const float* b3p = W_ih + (size_t)(n0 + 48 + l) * EDIM + 2 * half;

  for (int k0 = 0; k0 < EDIM; k0 += 4) {
    v2f a  = *(const v2f*)(arow + k0);
    v2f b0 = *(const v2f*)(b0p + k0);
    v2f b1 = *(const v2f*)(b1p + k0);
    v2f b2 = *(const v2f*)(b2p + k0);
    v2f b3 = *(const v2f*)(b3p + k0);
    acc0 = __builtin_amdgcn_wmma_f32_16x16x4_f32(false, a, false, b0, (short)0, acc0, false, false);
    acc1 = __builtin_amdgcn_wmma_f32_16x16x4_f32(false, a, false, b1, (short)0, acc1, false, false);
    acc2 = __builtin_amdgcn_wmma_f32_16x16x4_f32(false, a, false, b2, (short)0, acc2, false, false);
    acc3 = __builtin_amdgcn_wmma_f32_16x16x4_f32(false, a, false, b3, (short)0, acc3, false, false);
  }

  // Store with bias, time-major layout xp[t][b][h].
  v8f accs[4] = {acc0, acc1, acc2, acc3};
#pragma unroll
  for (int j = 0; j < 4; ++j) {
    const int col  = n0 + j * 16 + l;
    const float bias = b_ih[col] + b_hh[col];
#pragma unroll
    for (int r = 0; r < 8; ++r) {
      const int mr = m0 + half * 8 + r;      // flat b*T + t
      const int t  = mr & (TDIM - 1);
      const int b  = mr >> 9;                // T = 512
      xp[((size_t)t * BDIM + b) * HDIM + col] = accs[j][r] + bias;
    }
  }
}

#if HAVE_TDM
// Issue a TDM load of a 2-D tile: 128 rows x 32 floats, row stride 512 floats,
// from gsrc into LDS at byte offset lds_off (compact 32-float rows).
// D# per CDNA5 ISA ch.8: group0 = {count|flags, lds_addr, global_addr, type},
// group1 = dims/strides. Trailing groups zeroed (<=2-D tensor, count=1).
__device__ __forceinline__ void tdm_load_xp_tile(const float* gsrc, unsigned lds_off) {
  unsigned long long ga = (unsigned long long)(const void*)gsrc;
  v4u g0;
  g0.x = 1u;                                        // count=1 (valid user D#)
  g0.y = lds_off;                                   // lds_addr (bytes)
  g0.z = (unsigned)(ga & 0xffffffffu);              // global_addr[31:0]
  g0.w = (unsigned)((ga >> 32) & 0x01ffffffu)       // global_addr[56:32]
       | (2u << 30);                                // type=2 ("image")
  v8i g1;
  g1[0] = (int)(2u << 16);        // workgroup_mask=0 | data_size=2 (4B) | no pad
  g1[1] = (int)(512u << 16);      // atomic_barrier_addr=0 | tensor_dim0.lo=512
  g1[2] = (int)(128u << 16);      // tensor_dim0.hi=0     | tensor_dim1.lo=128
  g1[3] = (int)(32u  << 16);      // tensor_dim1.hi=0     | tile_dim0=32
  g1[4] = (int)(128u);            // tile_dim1=128        | tile_dim2=0
  g1[5] = (int)(512u);            // tensor_dim0_stride[31:0]=512
  g1[6] = 0;                      // stride.hi=0 | tensor_dim1_stride.lo=0
  g1[7] = 0;
  v4i gz4 = {0, 0, 0, 0};
  v8i gz8 = {0, 0, 0, 0, 0, 0, 0, 0};
  __builtin_amdgcn_tensor_load_to_lds(g0, g1, gz4, gz4, gz8, 0);
}
#endif

// ---------------------------------------------------------------------------
// Kernel 2: sequential recurrence, persistent grid with device-scope barrier.
//   h_{t+1} = tanh(xp_t + h_t @ W_hh^T)
// 16 blocks x 256 threads (8 waves). Block j owns h columns [32j, 32j+32).
//  - W_hh slice (32x512, 64 KB) staged to LDS once (async-to-LDS path).
//  - xp_t panel (128x32, 16 KB) double-buffered in LDS via the Tensor Data
//    Mover, issued one step ahead; s_wait_tensorcnt before the grid barrier.
// ---------------------------------------------------------------------------
__global__ __launch_bounds__(256)
void rnn_recurrence_kernel(const float* __restrict__ xp,
                           const float* __restrict__ W_hh,
                           float* __restrict__ hbuf0,
                           float* __restrict__ hbuf1,
                           unsigned* __restrict__ counter) {
  __shared__ float wsh[32 * WPAD];          // 66 KB, padded rows
  __shared__ float xps[2][BDIM * 32];       // 2 x 16 KB ping-pong xp panels

  const int lane = threadIdx.x & 31;
  const int wave = threadIdx.x >> 5;
  const int l    = lane & 15;
  const int half = lane >> 4;

  const int m0    = wave * 16;
  const int ncol0 = blockIdx.x * 32;

  // ---- stage W_hh[ncol0 .. ncol0+31][0..511] into LDS (once) ----
  for (int idx = threadIdx.x; idx < 32 * 128; idx += 256) {
    const int r = idx >> 7;
    const int c = (idx & 127) * 4;
    const float* g = W_hh + (size_t)(ncol0 + r) * HDIM + c;
#if HAVE_ASYNC_LDS
    __builtin_amdgcn_global_load_async_to_lds_b128(
        (v4i*)const_cast<float*>(g), (v4i*)&wsh[r * WPAD + c], 0, 0);
#else
    v2f lo = *(const v2f*)(g + 0);
    v2f hi = *(const v2f*)(g + 2);
    *(v2f*)&wsh[r * WPAD + c + 0] = lo;
    *(v2f*)&wsh[r * WPAD + c + 2] = hi;
#endif
  }

  // ---- preload xp panel for t=0 into buffer 0 ----
#if HAVE_TDM
  if (wave == 0) {
    tdm_load_xp_tile(xp + (size_t)0 * BDIM * HDIM + ncol0,
                     (unsigned)(unsigned long long)(void*)&xps[0][0]);
    __builtin_amdgcn_s_wait_tensorcnt(0);
  }
#else
  for (int idx = threadIdx.x; idx < BDIM * 8; idx += 256) {
    const int r = idx >> 3;
    const int c = (idx & 7) * 4;
    const float* g = xp + (size_t)r * HDIM + ncol0 + c;
#if HAVE_ASYNC_LDS
    __builtin_amdgcn_global_load_async_to_lds_b128(
        (v4i*)const_cast<float*>(g), (v4i*)&xps[0][r * 32 + c], 0, 0);
#else
    v2f lo = *(const v2f*)(g + 0);
    v2f hi = *(const v2f*)(g + 2);
    *(v2f*)&xps[0][r * 32 + c + 0] = lo;
    *(v2f*)&xps[0][r * 32 + c + 2] = hi;
#endif
  }
#endif
#if HAVE_ASYNC_LDS
#if __has_builtin(__builtin_amdgcn_s_wait_asynccnt)
  __builtin_amdgcn_s_wait_asynccnt(0);
#else
  asm volatile("s_wait_asynccnt 0" ::: "memory");
#endif
#endif
  __syncthreads();

  // LDS B-fragment base pointers (local rows jt*16 + l)
  const float* w0s = &wsh[(size_t)( 0 + l) * WPAD + 2 * half];
  const float* w1s = &wsh[(size_t)(16 + l) * WPAD + 2 * half];

  const unsigned nblk = gridDim.x;

  for (int t = 0; t < TDIM; ++t) {
    const float* hr = (t & 1) ? hbuf1 : hbuf0;
    float*       hw = (t & 1) ? hbuf0 : hbuf1;
    const int p = t & 1;

    // kick off next timestep's xp panel into the other LDS buffer
    if (t + 1 < TDIM) {
#if HAVE_TDM
      if (wave == 0) {
        tdm_load_xp_tile(xp + (size_t)(t + 1) * BDIM * HDIM + ncol0,
                         (unsigned)(unsigned long long)(void*)&xps[1 - p][0]);
      }
#else
      for (int idx = threadIdx.x; idx < BDIM * 8; idx += 256) {
        const int r = idx >> 3;
        const int c = (idx & 7) * 4;
        const float* g = xp + (size_t)(t + 1) * BDIM * HDIM + (size_t)r * HDIM + ncol0 + c;
#if HAVE_ASYNC_LDS
        __builtin_amdgcn_global_load_async_to_lds_b128(
            (v4i*)const_cast<float*>(g), (v4i*)&xps[1 - p][r * 32 + c], 0, 0);
#else
        v2f lo = *(const v2f*)(g + 0);
        v2f hi = *(const v2f*)(g + 2);
        *(v2f*)&xps[1 - p][r * 32 + c + 0] = lo;
        *(v2f*)&xps[1 - p][r * 32 + c + 2] = hi;
#endif
      }
#endif
    }

    v8f acc0 = {}, acc1 = {};
    const float* ap = hr + (size_t)(m0 + l) * HDIM + 2 * half;

#pragma unroll 4
    for (int k0 = 0; k0 < HDIM; k0 += 4) {
      v2f a  = *(const v2f*)(ap + k0);
      v2f b0 = *(const v2f*)(w0s + k0);   // ds_load_b64, bank-conflict-free
      v2f b1 = *(const v2f*)(w1s + k0);
      acc0 = __builtin_amdgcn_wmma_f32_16x16x4_f32(false, a, false, b0, (short)0, acc0, false, false);
      acc1 = __builtin_amdgcn_wmma_f32_16x16x4_f32(false, a, false, b1, (short)0, acc1, false, false);
    }

    // tail: add xp (from LDS panel), tanh, write h_new
    const float* xpl = &xps[p][0];
    v8f accs[2] = {acc0, acc1};
#pragma unroll
    for (int j = 0; j < 2; ++j) {
      const int col  = ncol0 + j * 16 + l;   // global column
      const int lcol = j * 16 + l;           // column within LDS panel
#pragma unroll
      for (int r = 0; r < 8; ++r) {
        const int row = m0 + half * 8 + r;
        float v = tanhf(accs[j][r] + xpl[row * 32 + lcol]);
        hw[(size_t)row * HDIM + col] = v;
      }
    }

    // make sure next panel's DMA (and fallback async loads) have landed
#if HAVE_TDM
    if (wave == 0) __builtin_amdgcn_s_wait_tensorcnt(0);
#elif HAVE_ASYNC_LDS
#if __has_builtin(__builtin_amdgcn_s_wait_asynccnt)
    __builtin_amdgcn_s_wait_asynccnt(0);
#else
    asm volatile("s_wait_asynccnt 0" ::: "memory");
#endif
#endif

    // ---- device-scope grid barrier (monotonic counter) ----
    __threadfence();
    __syncthreads();
    if (threadIdx.x == 0) {
      __hip_atomic_fetch_add(counter, 1u, __ATOMIC_RELEASE, __HIP_MEMORY_SCOPE_AGENT);
      const unsigned target = (unsigned)(t + 1) * nblk;
      while (__hip_atomic_load(counter, __ATOMIC_ACQUIRE, __HIP_MEMORY_SCOPE_AGENT) < target) {
        __builtin_amdgcn_s_sleep(2);
      }
    }
    __syncthreads();
  }
}

// ---------------------------------------------------------------------------
// Kernel 3: tiny FC head: y[b][o] = h[b] . W_fc[o] + b_fc[o]
// ---------------------------------------------------------------------------
__global__ __launch_bounds__(256)
void fc_kernel(const float* __restrict__ h,
               const float* __restrict__ W_fc,
               const float* __restrict__ b_fc,
               float* __restrict__ y) {
  int i = blockIdx.x * blockDim.x + threadIdx.x;
  if (i >= BDIM * ODIM) return;
  const int b = i >> 2;
  const int o = i & 3;
  const float* hp = h + (size_t)b * HDIM;
  const float* wp = W_fc + (size_t)o * HDIM;
  float s = b_fc[o];
#pragma unroll 4
  for (int k = 0; k < HDIM; ++k) s += hp[k] * wp[k];
  y[i] = s;
}

// ---------------------------------------------------------------------------
extern "C" void kernel_launch(void* const* d_in, const int* in_sizes, int n_in,
                              void* d_out, int out_size, void* d_ws, size_t ws_size,
                              hipStream_t stream) {
  const int*   x    = (const int*)  d_in[0];
  const float* emb  = (const float*)d_in[1];
  const float* W_ih = (const float*)d_in[2];
  const float* W_hh = (const float*)d_in[3];
  const float* b_ih = (const float*)d_in[4];
  const float* b_hh = (const float*)d_in[5];
  const float* W_fc = (const float*)d_in[6];
  const float* b_fc = (const float*)d_in[7];
  float* y = (float*)d_out;

  // workspace layout (floats): xp [T*B*H] | h0 [B*H] | h1 [B*H] | counter
  float* ws       = (float*)d_ws;
  const size_t xpElems = (size_t)TDIM * BDIM * HDIM;   // 33,554,432
  const size_t hElems  = (size_t)BDIM * HDIM;          // 65,536
  float* xp    = ws;
  float* h0    = ws + xpElems;
  float* h1    = h0 + hElems;
  unsigned* counter = (unsigned*)(h1 + hElems);

  // 1) init h0 = 0, counter = 0 (required every launch; harness poisons ws once)
  {
    const int n = (int)hElems;
    rnn_init_kernel<<<(n + 255) / 256, 256, 0, stream>>>(h0, counter, n);
  }

  // 2) embedding gather + input projection: grid (M/64, H/64)
  {
    dim3 grid((BDIM * TDIM) / 64, HDIM / 64);
    embed_proj_kernel<<<grid, 128, 0, stream>>>(x, emb, W_ih, b_ih, b_hh, xp);
  }

  // 3) sequential recurrence: 16 persistent blocks, final h lands in h0
  rnn_recurrence_kernel<<<16, 256, 0, stream>>>(xp, W_hh, h0, h1, counter);

  // 4) FC head
  fc_kernel<<<2, 256, 0, stream>>>(h0, W_fc, b_fc, y);
}